// RingConv2dSimple_61856118997345
// MI455X (gfx1250) — compile-verified
//
#include <hip/hip_runtime.h>
#include <math.h>

// RingConv2d on MI455X (gfx1250, wave32).
// cos/sin angle-difference identity turns the ring conv into 4 accumulating
// GEMMs: M=32 (cout), K=144 (cin*3*3), N=16384 (b*ho*wo), then atan2.
//   dirX = Wc.cx + Ws.sx ;  dirY = Wc.sx - Ws.cx
// Full-precision V_WMMA_F32_16X16X4_F32 (problem is HBM/transcendental bound:
// 3 MB traffic ~0.13us @23.3TB/s, 302M MACs is trivial for the WMMA units).

typedef float v2f __attribute__((ext_vector_type(2)));
typedef float v8f __attribute__((ext_vector_type(8)));
typedef int   v2i __attribute__((ext_vector_type(2)));

#define CIN   16
#define COUT  32
#define HH    64
#define WW    64
#define RTOT  144          // CIN * 3 * 3
#define LCOLS 66           // padded columns: input col -1..64

__global__ __launch_bounds__(128)
void ringconv_wmma_kernel(const float* __restrict__ x,
                          const float* __restrict__ wgt,
                          float* __restrict__ out)
{
    // LDS staging. Weights stored K-paired: w[(r>>1)*64 + co*2 + (r&1)] so an
    // A-fragment (elements r0, r0+1 for one co) is one aligned ds_load_b64.
    __shared__ __align__(16) float cxT[CIN * 3 * LCOLS];   // cos(x) band
    __shared__ __align__(16) float sxT[CIN * 3 * LCOLS];   // sin(x) band
    __shared__ __align__(16) float wcP[RTOT * COUT];       // cos(w), paired
    __shared__ __align__(16) float wsP[RTOT * COUT];       // sin(w), paired
    __shared__ __align__(16) int   offT[RTOT];             // r -> band offset

    const int tid  = threadIdx.x;
    const int lane = tid & 31;
    const int wave = tid >> 5;
    const int b    = blockIdx.x >> 6;        // 4 batches
    const int ho   = blockIdx.x & 63;        // 64 output rows

    // ---- stage input band: rows ho-1..ho+1, cols -1..64, all 16 cin ----
    for (int idx = tid; idx < CIN * 3 * LCOLS; idx += 128) {
        int ci  = idx / (3 * LCOLS);
        int rem = idx - ci * 3 * LCOLS;
        int j   = rem / LCOLS;               // 0..2 -> row ho-1+j
        int c   = rem - j * LCOLS;           // 0..65 -> col c-1
        int row = ho - 1 + j;
        int col = c - 1;
        float cv = 1.0f, sv = 0.0f;          // zero-pad: cos(0)=1, sin(0)=0
        if (row >= 0 && row < HH && col >= 0 && col < WW) {
            float v = x[((b * CIN + ci) * HH + row) * WW + col];
            sincosf(v, &sv, &cv);
        }
        cxT[idx] = cv;
        sxT[idx] = sv;
    }

    // ---- stage weights (layout (1,cin,cout,1,1,3,3)), K-paired in LDS ----
    for (int idx = tid; idx < RTOT * COUT; idx += 128) {
        int r  = idx >> 5;                   // 0..143 = ci*9 + kh*3 + kw
        int co = idx & 31;
        int ci = r / 9;
        int rr = r - ci * 9;
        float v = wgt[(ci * COUT + co) * 9 + rr];
        float cv, sv;
        sincosf(v, &sv, &cv);
        int pi = (r >> 1) * (COUT * 2) + co * 2 + (r & 1);
        wcP[pi] = cv;
        wsP[pi] = sv;
    }

    // ---- im2col offset table: r -> (ci*3+kh)*LCOLS + kw ----
    for (int r = tid; r < RTOT; r += 128) {
        int ci = r / 9;
        int rr = r - ci * 9;
        int kh = rr / 3;
        int kw = rr - kh * 3;
        offT[r] = (ci * 3 + kh) * LCOLS + kw;
    }
    __syncthreads();

    // ---- WMMA fragment coordinates (f32 16x16x4 layout) ----
    // A (16x4): lanes 0-15 M=lane, K={k0,k0+1}; lanes 16-31 M=lane-16, K={k0+2,k0+3}
    // B (4x16): same K split, N=lane&15.
    const int ml   = lane & 15;              // co_local for A, n for B
    const int khi  = lane >> 4;              // 0/1 -> K pair select
    const int bcol = wave * 16 + ml;         // wo for this lane's B column

    v8f accX0 = {}, accYa0 = {}, accYb0 = {};   // cout 0..15
    v8f accX1 = {}, accYa1 = {}, accYb1 = {};   // cout 16..31

    #pragma unroll 4
    for (int k0 = 0; k0 < RTOT; k0 += 4) {
        const int p  = (k0 >> 1) + khi;      // K-pair index (r0 = 2p, even)
        const int r0 = 2 * p;

        // B fragments: gathered im2col of cos/sin input via offset table
        v2i oo = *(const v2i*)&offT[r0];     // one ds_load_b64
        v2f bc, bs;
        bc.x = cxT[oo.x + bcol]; bc.y = cxT[oo.y + bcol];
        bs.x = sxT[oo.x + bcol]; bs.y = sxT[oo.y + bcol];

        // A fragments: one ds_load_b64 each (paired layout)
        const int wbase = p * (COUT * 2) + ml * 2;
        v2f ac0 = *(const v2f*)&wcP[wbase];
        v2f as0 = *(const v2f*)&wsP[wbase];
        v2f ac1 = *(const v2f*)&wcP[wbase + 32];
        v2f as1 = *(const v2f*)&wsP[wbase + 32];

        // accX = Wc.cx + Ws.sx ; accYa = Wc.sx ; accYb = Ws.cx
        accX0  = __builtin_amdgcn_wmma_f32_16x16x4_f32(false, ac0, false, bc, (short)0, accX0,  false, false);
        accX0  = __builtin_amdgcn_wmma_f32_16x16x4_f32(false, as0, false, bs, (short)0, accX0,  false, false);
        accYa0 = __builtin_amdgcn_wmma_f32_16x16x4_f32(false, ac0, false, bs, (short)0, accYa0, false, false);
        accYb0 = __builtin_amdgcn_wmma_f32_16x16x4_f32(false, as0, false, bc, (short)0, accYb0, false, false);

        accX1  = __builtin_amdgcn_wmma_f32_16x16x4_f32(false, ac1, false, bc, (short)0, accX1,  false, false);
        accX1  = __builtin_amdgcn_wmma_f32_16x16x4_f32(false, as1, false, bs, (short)0, accX1,  false, false);
        accYa1 = __builtin_amdgcn_wmma_f32_16x16x4_f32(false, ac1, false, bs, (short)0, accYa1, false, false);
        accYb1 = __builtin_amdgcn_wmma_f32_16x16x4_f32(false, as1, false, bc, (short)0, accYb1, false, false);
    }

    // ---- epilogue: dirY = accYa - accYb, atan2, store ----
    // C/D layout: VGPR i, lanes 0-15 -> M=i, lanes 16-31 -> M=i+8; N=lane&15.
    const int wo    = wave * 16 + ml;
    const int mhalf = khi * 8;
    #pragma unroll
    for (int i = 0; i < 8; ++i) {
        int co0 = i + mhalf;
        int co1 = co0 + 16;
        float y0 = accYa0[i] - accYb0[i];
        float y1 = accYa1[i] - accYb1[i];
        out[((b * COUT + co0) * HH + ho) * WW + wo] = atan2f(y0, accX0[i]);
        out[((b * COUT + co1) * HH + ho) * WW + wo] = atan2f(y1, accX1[i]);
    }
}

extern "C" void kernel_launch(void* const* d_in, const int* in_sizes, int n_in,
                              void* d_out, int out_size, void* d_ws, size_t ws_size,
                              hipStream_t stream) {
    (void)in_sizes; (void)n_in; (void)out_size; (void)d_ws; (void)ws_size;
    const float* x   = (const float*)d_in[0];   // (4,16,64,64)
    const float* wgt = (const float*)d_in[1];   // (1,16,32,1,1,3,3)
    float* out = (float*)d_out;                 // (4,32,64,64)
    dim3 grid(4 * 64);      // one block per (b, ho) row
    dim3 block(128);        // 4 waves; each wave = 16 wo cols x all 32 cout
    ringconv_wmma_kernel<<<grid, block, 0, stream>>>(x, wgt, out);
}